// SqueezeExcitation_53274774339948
// MI455X (gfx1250) — compile-verified
//
#include <hip/hip_runtime.h>
#include <hip/hip_bf16.h>

typedef __attribute__((ext_vector_type(2))) float v2f;
typedef __attribute__((ext_vector_type(8))) float v8f;

#define HW   3136            // 56*56
#define HW4  784             // HW/4 float4 per (b,c) row
#define B_   64
#define CIN  256
#define CMID 64
#define NROW (B_ * CIN)      // 16384 (b,c) rows

// ---------------------------------------------------------------------------
// Kernel 1: global average pool.  One 256-thread block per (b,c) row.
// 128-bit loads, wave32 shuffle reduce, 8-wave LDS combine.
// ---------------------------------------------------------------------------
__global__ void se_pool_kernel(const float* __restrict__ x, float* __restrict__ s) {
    const int row = blockIdx.x;                       // 0..16383 == b*256 + c
    const float4* xr = (const float4*)(x + (size_t)row * HW);

    float sum = 0.0f;
    for (int i = threadIdx.x; i < HW4; i += 256) {    // 784 = 3*256 + 16
        float4 v = xr[i];
        sum += v.x + v.y + v.z + v.w;
    }
    // wave32 reduction
    #pragma unroll
    for (int off = 16; off > 0; off >>= 1)
        sum += __shfl_down(sum, off, 32);

    __shared__ float wsum[8];
    const int lane = threadIdx.x & 31;
    const int wave = threadIdx.x >> 5;
    if (lane == 0) wsum[wave] = sum;
    __syncthreads();
    if (threadIdx.x == 0) {
        float t = 0.0f;
        #pragma unroll
        for (int i = 0; i < 8; ++i) t += wsum[i];
        s[row] = t * (1.0f / (float)HW);
    }
}

// ---------------------------------------------------------------------------
// Kernel 2: SE MLP via V_WMMA_F32_16X16X4_F32.  One block, 512 threads = 16
// waves.  fc1: each wave owns one 16x16 tile of [64 x 64].  fc2: each wave
// owns four 16x16 tiles of [64 x 256].  No divergence -> EXEC all-1s.
// ---------------------------------------------------------------------------
__global__ void __launch_bounds__(512)
se_mlp_kernel(const float* __restrict__ s,  const float* __restrict__ w1,
              const float* __restrict__ b1, const float* __restrict__ w2,
              const float* __restrict__ b2, float* __restrict__ scale) {
    __shared__ float h[B_][CMID + 1];   // stride 65 -> conflict-free column reads

    const int lane  = threadIdx.x & 31;
    const int wave  = threadIdx.x >> 5;   // 0..15
    const int row   = lane & 15;          // M (A-load) or N (B-load / D col)
    const int khalf = lane >> 4;          // 0: K=k0,k0+1   1: K=k0+2,k0+3

    // ---- fc1: out1[b,m] = sum_c s[b,c] * w1[m,c] ----
    {
        const int bt = wave >> 2;         // batch tile  0..3
        const int mt = wave & 3;          // cmid tile   0..3
        v8f acc = {};
        const float* arow = s  + (size_t)(bt * 16 + row) * CIN;  // A: row = b
        const float* brow = w1 + (size_t)(mt * 16 + row) * CIN;  // B[c][m]=w1[m][c]
        for (int k0 = 0; k0 < CIN; k0 += 4) {
            const int ka = k0 + 2 * khalf;
            v2f a = { arow[ka], arow[ka + 1] };
            v2f b = { brow[ka], brow[ka + 1] };
            acc = __builtin_amdgcn_wmma_f32_16x16x4_f32(
                      false, a, false, b, (short)0, acc, false, false);
        }
        // D layout: VGPR i -> M = i + 8*khalf, N = row
        const int mm = (wave & 3) * 16 + row;
        const float bias = b1[mm];
        #pragma unroll
        for (int i = 0; i < 8; ++i) {
            const int bb = bt * 16 + i + 8 * khalf;
            float v = acc[i] + bias;
            h[bb][mm] = v > 0.0f ? v : 0.0f;          // ReLU
        }
    }
    __syncthreads();

    // ---- fc2: out2[b,c] = sum_m h[b,m] * w2[c,m], then hardsigmoid ----
    #pragma unroll
    for (int t = 0; t < 4; ++t) {
        const int tid2 = wave * 4 + t;    // 0..63 tiles
        const int bt2  = tid2 >> 4;       // batch tile 0..3
        const int ct   = tid2 & 15;       // cout tile  0..15
        v8f acc = {};
        const float* brow = w2 + (size_t)(ct * 16 + row) * CMID; // B[m][c]=w2[c][m]
        for (int k0 = 0; k0 < CMID; k0 += 4) {
            const int ka = k0 + 2 * khalf;
            v2f a = { h[bt2 * 16 + row][ka], h[bt2 * 16 + row][ka + 1] };
            v2f b = { brow[ka], brow[ka + 1] };
            acc = __builtin_amdgcn_wmma_f32_16x16x4_f32(
                      false, a, false, b, (short)0, acc, false, false);
        }
        const int cc = ct * 16 + row;
        const float bias = b2[cc];
        #pragma unroll
        for (int i = 0; i < 8; ++i) {
            const int bb = bt2 * 16 + i + 8 * khalf;
            float v = (acc[i] + bias + 3.0f) * (1.0f / 6.0f);   // hardsigmoid
            v = fminf(fmaxf(v, 0.0f), 1.0f);
            scale[bb * CIN + cc] = v;
        }
    }
}

// ---------------------------------------------------------------------------
// Kernel 3: out = x * scale[b,c], pure float4 stream (HBM-bound).
// 12,845,056 float4 total = 50176 blocks * 256 threads exactly.
// ---------------------------------------------------------------------------
__global__ void se_scale_kernel(const float* __restrict__ x,
                                const float* __restrict__ sc,
                                float* __restrict__ out) {
    const unsigned idx = blockIdx.x * 256u + threadIdx.x;   // float4 index
    const unsigned row = idx / HW4;                          // b*256 + c
    const float sv = sc[row];
    float4 v = ((const float4*)x)[idx];
    v.x *= sv; v.y *= sv; v.z *= sv; v.w *= sv;
    ((float4*)out)[idx] = v;
}

// ---------------------------------------------------------------------------
extern "C" void kernel_launch(void* const* d_in, const int* in_sizes, int n_in,
                              void* d_out, int out_size, void* d_ws, size_t ws_size,
                              hipStream_t stream) {
    const float* x  = (const float*)d_in[0];
    const float* w1 = (const float*)d_in[1];
    const float* b1 = (const float*)d_in[2];
    const float* w2 = (const float*)d_in[3];
    const float* b2 = (const float*)d_in[4];
    float* out = (float*)d_out;

    float* s_buf  = (float*)d_ws;            // [64*256] pooled means
    float* sc_buf = s_buf + NROW;            // [64*256] SE scales

    se_pool_kernel<<<NROW, 256, 0, stream>>>(x, s_buf);
    se_mlp_kernel<<<1, 512, 0, stream>>>(s_buf, w1, b1, w2, b2, sc_buf);

    const unsigned n4 = (unsigned)((size_t)B_ * CIN * HW / 4);   // 12,845,056
    se_scale_kernel<<<n4 / 256, 256, 0, stream>>>(x, sc_buf, out);
}